// Neigh_Agg_49323404427453
// MI455X (gfx1250) — compile-verified
//
#include <hip/hip_runtime.h>

#define DIM 64

typedef __attribute__((ext_vector_type(2))) float v2f;
typedef __attribute__((ext_vector_type(8))) float v8f;

// ---------------------------------------------------------------------------
// Zero the output accumulator and the per-node counts.
// ---------------------------------------------------------------------------
__global__ void zero_kernel(float* __restrict__ out, float* __restrict__ counts,
                            int n_nodes) {
    int i = blockIdx.x * blockDim.x + threadIdx.x;
    if (i < n_nodes * DIM) out[i] = 0.0f;
    if (i < n_nodes) counts[i] = 0.0f;
}

// ---------------------------------------------------------------------------
// h = relu(x @ W^T + b) using V_WMMA_F32_16X16X4_F32 (full fp32 precision).
// One wave computes a 16-node x 64-output tile.
//   A fragment (16x4 f32, 2 VGPRs): lane m=l&15 is row M; lane bit4 selects
//     the K pair {k0+2*hi, k0+2*hi+1}.
//   B fragment (4x16 f32, 2 VGPRs): lane m=l&15 is column N (output channel);
//     same K-pair selection. B[k][n] = W[n][k] (Linear weight is [out,in]).
//   C/D (16x16 f32, 8 VGPRs): element r <-> row r+8*hi, column m.
// Out-of-range rows are CLAMPED (not masked) so EXEC stays all-1s through the
// entire K loop: no divergence, no saveexec, clean load+WMMA pipeline. The
// epilogue store guard discards the duplicated rows.
// ---------------------------------------------------------------------------
__global__ void gemm_relu_kernel(const float* __restrict__ x,
                                 const float* __restrict__ W,
                                 const float* __restrict__ bias,
                                 float* __restrict__ h, int n_nodes) {
    int wave  = (blockIdx.x * blockDim.x + threadIdx.x) >> 5;
    int lane  = threadIdx.x & 31;
    int node0 = wave * 16;
    if (node0 >= n_nodes) return;           // uniform across the wave

    int m     = lane & 15;                  // A row / B,D column within tile
    int hi    = lane >> 4;                  // which K-pair / which row half
    int khalf = hi * 2;

    int row = node0 + m;
    if (row > n_nodes - 1) row = n_nodes - 1;   // clamp: keep EXEC all-1s

    const float* xrow = x + (size_t)row * DIM + khalf;   // lane's A base
    const float* wrow = W + (size_t)m * DIM + khalf;     // lane's B base (ot=0)

    v8f acc[4];
    #pragma unroll
    for (int ot = 0; ot < 4; ++ot)
        acc[ot] = (v8f){0.f,0.f,0.f,0.f,0.f,0.f,0.f,0.f};

    #pragma unroll
    for (int k0 = 0; k0 < DIM; k0 += 4) {
        v2f a;
        a.x = xrow[k0 + 0];
        a.y = xrow[k0 + 1];
        #pragma unroll
        for (int ot = 0; ot < 4; ++ot) {
            v2f bb;
            bb.x = wrow[ot * 16 * DIM + k0 + 0];
            bb.y = wrow[ot * 16 * DIM + k0 + 1];
            // 8 args: (neg_a, A, neg_b, B, c_mod, C, reuse_a, reuse_b)
            acc[ot] = __builtin_amdgcn_wmma_f32_16x16x4_f32(
                false, a, false, bb, (short)0, acc[ot], false, false);
        }
    }

    // Epilogue: bias + ReLU + store (guard handles any tail tile).
    #pragma unroll
    for (int ot = 0; ot < 4; ++ot) {
        int o  = ot * 16 + m;
        float bv = bias[o];
        #pragma unroll
        for (int r = 0; r < 8; ++r) {
            int rrow = node0 + r + 8 * hi;
            if (rrow < n_nodes) {
                float v = acc[ot][r] + bv;
                h[(size_t)rrow * DIM + o] = v > 0.0f ? v : 0.0f;
            }
        }
    }
}

// ---------------------------------------------------------------------------
// Per-edge gather of h[target] and atomic scatter-add into out[source].
// 16 threads per edge; each thread moves 4 consecutive floats (float4 gather,
// four global_atomic_add_f32). Lane 0 of each edge bumps the source count
// (float accumulation is exact: max count 1.6M < 2^24).
// Both h (25.6 MB) and out (25.6 MB) are L2-resident (192 MB L2), so this
// phase runs at L2 gather/atomic throughput — it dominates total runtime.
// ---------------------------------------------------------------------------
__global__ void scatter_edges_kernel(const int* __restrict__ ei,   // [2, E]
                                     const float* __restrict__ h,
                                     float* __restrict__ out,
                                     float* __restrict__ counts,
                                     int n_edges) {
    long long t = (long long)blockIdx.x * blockDim.x + threadIdx.x;
    int e   = (int)(t >> 4);
    int sub = (int)(t & 15);
    if (e >= n_edges) return;

    int s = ei[e];              // source node (row 0)
    int g = ei[n_edges + e];    // target node (row 1)

    const float4* hp = (const float4*)(h + (size_t)g * DIM);
    float4 v = hp[sub];

    float* dst = out + (size_t)s * DIM + sub * 4;
    atomicAdd(dst + 0, v.x);
    atomicAdd(dst + 1, v.y);
    atomicAdd(dst + 2, v.z);
    atomicAdd(dst + 3, v.w);

    if (sub == 0) atomicAdd(counts + s, 1.0f);
}

// ---------------------------------------------------------------------------
// out[n][d] /= max(counts[n], 1)
// ---------------------------------------------------------------------------
__global__ void normalize_kernel(float* __restrict__ out,
                                 const float* __restrict__ counts,
                                 int n_nodes) {
    int i = blockIdx.x * blockDim.x + threadIdx.x;
    if (i >= n_nodes * DIM) return;
    float c = counts[i >> 6];
    c = c > 1.0f ? c : 1.0f;
    out[i] = out[i] / c;
}

extern "C" void kernel_launch(void* const* d_in, const int* in_sizes, int n_in,
                              void* d_out, int out_size, void* d_ws, size_t ws_size,
                              hipStream_t stream) {
    const float* x  = (const float*)d_in[0];
    const int*   ei = (const int*)d_in[1];
    const float* W  = (const float*)d_in[2];
    const float* b  = (const float*)d_in[3];

    int n_nodes = in_sizes[0] / DIM;
    int n_edges = in_sizes[1] / 2;

    float* out    = (float*)d_out;
    float* h      = (float*)d_ws;                       // n_nodes*64 floats
    float* counts = h + (size_t)n_nodes * DIM;          // n_nodes floats

    // 1) zero accumulators
    int zn = n_nodes * DIM;
    zero_kernel<<<(zn + 255) / 256, 256, 0, stream>>>(out, counts, n_nodes);

    // 2) h = relu(x W^T + b): one wave per 16-node tile, 8 waves per block
    int ntiles  = (n_nodes + 15) / 16;
    int nblocks = (ntiles + 7) / 8;
    gemm_relu_kernel<<<nblocks, 256, 0, stream>>>(x, W, b, h, n_nodes);

    // 3) edge gather + atomic scatter-add (16 threads per edge)
    long long sth = (long long)n_edges * 16;
    scatter_edges_kernel<<<(int)((sth + 255) / 256), 256, 0, stream>>>(
        ei, h, out, counts, n_edges);

    // 4) mean normalization
    normalize_kernel<<<(zn + 255) / 256, 256, 0, stream>>>(out, counts, n_nodes);
}